// Masking_76424648065762
// MI455X (gfx1250) — compile-verified
//
#include <hip/hip_runtime.h>

// Problem shape (compile-time constants from the reference).
#define B_  8
#define L_  2048
#define C_  64
#define D_  32
#define N_  16

typedef __attribute__((ext_vector_type(4))) float v4f;

// One thread per (b, l, c) row: 32 floats = 128 bytes, moved as 8x b128.
// Mask test (16 blocks) amortized once per row; params are wave-uniform so
// they lower to scalar (s_load) broadcasts.
__global__ __launch_bounds__(256) void Masking_76424648065762_kernel(
    const float* __restrict__ x,
    const int*   __restrict__ starts,
    const int*   __restrict__ lengths,
    const int*   __restrict__ channels,
    float*       __restrict__ out)
{
    const int row = (int)(blockIdx.x * blockDim.x + threadIdx.x); // < 2^20
    const int c = row & (C_ - 1);          // channel index
    const int l = (row >> 6) & (L_ - 1);   // sequence position (C_ == 64)

    // keep = 0 if any of the 16 blocks covers (l, c); branchless v_cndmask chain.
    float keep = 1.0f;
#pragma unroll
    for (int n = 0; n < N_; ++n) {
        const int s   = starts[n];    // uniform -> s_load_b32
        const int len = lengths[n];   // uniform -> s_load_b32
        const int ch  = channels[n];  // uniform -> s_load_b32
        const bool hit = (ch == c) && (l >= s) && (l < s + len);
        keep = hit ? 0.0f : keep;
    }

    // Row base in units of 16-byte vectors: row * (D_/4). Fits in 32-bit,
    // so addressing stays SADDR(base) + 32-bit voffset.
    const int vbase = row * (D_ / 4);
    const v4f* __restrict__ src = reinterpret_cast<const v4f*>(x)   + vbase;
    v4f*       __restrict__ dst = reinterpret_cast<v4f*>(out)       + vbase;

#pragma unroll
    for (int i = 0; i < D_ / 4; ++i) {
        // Streaming data touched exactly once: non-temporal both ways so the
        // 256 MiB stream doesn't evict useful lines from WGP$/L2.
        v4f v = __builtin_nontemporal_load(&src[i]);
        v.x *= keep;
        v.y *= keep;
        v.z *= keep;
        v.w *= keep;
        __builtin_nontemporal_store(v, &dst[i]);
    }
}

extern "C" void kernel_launch(void* const* d_in, const int* in_sizes, int n_in,
                              void* d_out, int out_size, void* d_ws, size_t ws_size,
                              hipStream_t stream) {
    (void)in_sizes; (void)n_in; (void)out_size; (void)d_ws; (void)ws_size;

    const float* x        = (const float*)d_in[0];
    const int*   starts   = (const int*)  d_in[1];
    const int*   lengths  = (const int*)  d_in[2];
    const int*   channels = (const int*)  d_in[3];
    float*       out      = (float*)      d_out;

    // B*L*C = 1,048,576 rows; 256 threads/block (8 wave32) -> 4096 blocks.
    const int rows    = B_ * L_ * C_;
    const int threads = 256;
    const int blocks  = rows / threads; // exact: 4096

    Masking_76424648065762_kernel<<<blocks, threads, 0, stream>>>(
        x, starts, lengths, channels, out);
}